// myGCN_75900662054956
// MI455X (gfx1250) — compile-verified
//
#include <hip/hip_runtime.h>

// ---------------------------------------------------------------------------
// myGCN interaction scorer for MI455X (gfx1250, wave32).
//
// Structure: one block per batch row b (1024 blocks x 256 threads).
//  Phase 1: 16 threads build user_rep[16] in LDS (59 row gathers, trivial).
//  Phase 2: all 256 threads pool course_user (50 rows) + course_teacher (5)
//           per (b,c); 4 lanes per c, float4 per lane -> fully coalesced
//           64B-row gathers (tables are L2-resident: 6.9 MB total vs 192 MB).
//           Course reps staged in LDS [112 x 16] (rows 100..111 zeroed).
//  Phase 3: wave 0 computes out[b, 0..99] with V_WMMA_F32_16X16X4_F32:
//           A = 16 course-rep rows (M=c, K=d), B = user_rep broadcast over N,
//           4 chained K=4 WMMAs cover D=16; column N=0 of the 16x16 f32
//           accumulator (lanes 0 and 16, VGPRs 0..7) holds the 16 dots.
// ---------------------------------------------------------------------------

typedef __attribute__((ext_vector_type(2))) float v2f;
typedef __attribute__((ext_vector_type(8))) float v8f;

namespace {
constexpr int kB    = 1024;
constexpr int kC    = 100;
constexpr int kLSEQ = 50;
constexpr int kLUT  = 5;
constexpr int kLUS  = 3;
constexpr int kLU   = 50;
constexpr int kLCT  = 5;
constexpr int kD    = 16;
constexpr int kCPad = 112;   // 7 WMMA tiles of 16 rows
}

__global__ __launch_bounds__(256)
void myGCN_75900662054956_kernel(
    const int*   __restrict__ user_idx,          // [B]
    const int*   __restrict__ user_sequence,     // [B, LSEQ]
    const int*   __restrict__ user_teachers,     // [B, LUT]
    const int*   __restrict__ user_school,       // [B, LUS]
    const float* __restrict__ user_len_seq,      // [B, 1]
    const float* __restrict__ user_len_teacher,  // [B, 1]
    const float* __restrict__ user_len_school,   // [B, 1]
    const int*   __restrict__ course_set,        // [B, C]
    const int*   __restrict__ course_user,       // [B, C, LU]
    const int*   __restrict__ course_teacher,    // [B, C, LCT]
    const float* __restrict__ course_len_u,      // [B, C]
    const float* __restrict__ course_len_teacher,// [B, C]
    const float* __restrict__ user_table,        // [NU+1, D]
    const float* __restrict__ course_table,      // [NC+1, D]
    const float* __restrict__ teacher_table,     // [NT+1, D]
    const float* __restrict__ school_table,      // [NS+1, D]
    float*       __restrict__ out)               // [B, C]
{
  __shared__ __align__(16) float s_user[kD];
  __shared__ __align__(16) float s_crep[kCPad * kD];

  const int b   = blockIdx.x;
  const int tid = threadIdx.x;

  // Zero the padding rows (c = 100..111) so the partial WMMA tile is benign.
  if (tid < (kCPad - kC) * kD) s_crep[kC * kD + tid] = 0.0f;

  // ---------------- Phase 1: user representation ----------------
  if (tid < kD) {
    const int d = tid;
    float ue = user_table[(size_t)user_idx[b] * kD + d];
    float s = 0.0f;
    for (int l = 0; l < kLSEQ; ++l)
      s += course_table[(size_t)user_sequence[b * kLSEQ + l] * kD + d];
    float t = 0.0f;
    for (int l = 0; l < kLUT; ++l)
      t += teacher_table[(size_t)user_teachers[b * kLUT + l] * kD + d];
    float sc = 0.0f;
    for (int l = 0; l < kLUS; ++l)
      sc += school_table[(size_t)user_school[b * kLUS + l] * kD + d];
    s_user[d] = (s / user_len_seq[b] + t / user_len_teacher[b]
                 + sc / user_len_school[b] + ue) * (1.0f / 3.0f);
  }

  // ---------------- Phase 2: course representations ----------------
  // 256 threads = 64 c's per pass, 4 lanes per c, float4 (quarter row) each.
  for (int pass = 0; pass < 2; ++pass) {
    const int c = pass * 64 + (tid >> 2);
    if (c < kC) {
      const int dq = (tid & 3) * 4;
      const size_t bc = (size_t)b * kC + c;

      const int* __restrict__ cu = course_user + bc * kLU;
      float4 acc = make_float4(0.0f, 0.0f, 0.0f, 0.0f);
      #pragma unroll 5
      for (int l = 0; l < kLU; ++l) {
        const float4 e = *reinterpret_cast<const float4*>(
            user_table + (size_t)cu[l] * kD + dq);
        acc.x += e.x; acc.y += e.y; acc.z += e.z; acc.w += e.w;
      }

      const int* __restrict__ ct = course_teacher + bc * kLCT;
      float4 tac = make_float4(0.0f, 0.0f, 0.0f, 0.0f);
      #pragma unroll
      for (int l = 0; l < kLCT; ++l) {
        const float4 e = *reinterpret_cast<const float4*>(
            teacher_table + (size_t)ct[l] * kD + dq);
        tac.x += e.x; tac.y += e.y; tac.z += e.z; tac.w += e.w;
      }

      const float4 ce = *reinterpret_cast<const float4*>(
          course_table + (size_t)course_set[bc] * kD + dq);

      const float inv_lu = 1.0f / course_len_u[bc];
      const float inv_lt = 1.0f / course_len_teacher[bc];

      // (cu_mean + cu_sum[bug-faithful school term] + teacher_mean + emb) / 4
      float4 rep;
      rep.x = (acc.x * inv_lu + acc.x + tac.x * inv_lt + ce.x) * 0.25f;
      rep.y = (acc.y * inv_lu + acc.y + tac.y * inv_lt + ce.y) * 0.25f;
      rep.z = (acc.z * inv_lu + acc.z + tac.z * inv_lt + ce.z) * 0.25f;
      rep.w = (acc.w * inv_lu + acc.w + tac.w * inv_lt + ce.w) * 0.25f;

      *reinterpret_cast<float4*>(&s_crep[c * kD + dq]) = rep;
    }
  }

  __syncthreads();

  // ---------------- Phase 3: interaction dots via WMMA (wave 0) ----------------
  // V_WMMA_F32_16X16X4_F32 f32 A layout: lane L -> M = L%16,
  //   VGPR0 holds K = (L<16 ? 0 : 2), VGPR1 holds K = (L<16 ? 1 : 3).
  // B (4x16) mirrors with N = L%16; we broadcast user_rep across all N.
  if (tid < 32) {
    const int lane = tid;
    const int m    = lane & 15;          // row (course) within tile
    const int khi  = (lane >> 4) << 1;   // 0 for lanes 0-15, 2 for lanes 16-31

    v2f bf[4];
    #pragma unroll
    for (int kk = 0; kk < 4; ++kk)
      bf[kk] = *reinterpret_cast<const v2f*>(&s_user[kk * 4 + khi]);

    #pragma unroll
    for (int tile = 0; tile < 7; ++tile) {
      const int cbase = tile * 16;
      v8f acc = {};
      #pragma unroll
      for (int kk = 0; kk < 4; ++kk) {
        v2f a = *reinterpret_cast<const v2f*>(
            &s_crep[(cbase + m) * kD + kk * 4 + khi]);
        acc = __builtin_amdgcn_wmma_f32_16x16x4_f32(
            /*neg_a=*/false, a, /*neg_b=*/false, bf[kk],
            /*c_mod=*/(short)0, acc, /*reuse_a=*/false, /*reuse_b=*/false);
      }
      // C/D layout: VGPR v, lanes 0-15 -> (M=v,   N=lane),
      //                     lanes 16-31 -> (M=v+8, N=lane-16).
      // All N columns are identical (B broadcast); read column N=0.
      if (m == 0) {
        const int mbase = (lane >> 4) * 8;
        #pragma unroll
        for (int v = 0; v < 8; ++v) {
          const int c = cbase + mbase + v;
          if (c < kC) out[(size_t)b * kC + c] = acc[v];
        }
      }
    }
  }
}

extern "C" void kernel_launch(void* const* d_in, const int* in_sizes, int n_in,
                              void* d_out, int out_size, void* d_ws, size_t ws_size,
                              hipStream_t stream) {
  (void)in_sizes; (void)n_in; (void)out_size; (void)d_ws; (void)ws_size;

  const int*   user_idx           = (const int*)  d_in[0];
  const int*   user_sequence      = (const int*)  d_in[1];
  const int*   user_teachers      = (const int*)  d_in[2];
  const int*   user_school        = (const int*)  d_in[3];
  const float* user_len_seq       = (const float*)d_in[4];
  const float* user_len_teacher   = (const float*)d_in[5];
  const float* user_len_school    = (const float*)d_in[6];
  const int*   course_set         = (const int*)  d_in[7];
  const int*   course_user        = (const int*)  d_in[8];
  // d_in[9] = course_school: dead code in the reference (bug kept faithfully)
  const int*   course_teacher     = (const int*)  d_in[10];
  const float* course_len_u       = (const float*)d_in[11];
  const float* course_len_teacher = (const float*)d_in[12];
  const float* user_table         = (const float*)d_in[13];
  const float* course_table       = (const float*)d_in[14];
  const float* teacher_table      = (const float*)d_in[15];
  const float* school_table       = (const float*)d_in[16];

  myGCN_75900662054956_kernel<<<kB, 256, 0, stream>>>(
      user_idx, user_sequence, user_teachers, user_school,
      user_len_seq, user_len_teacher, user_len_school,
      course_set, course_user, course_teacher,
      course_len_u, course_len_teacher,
      user_table, course_table, teacher_table, school_table,
      (float*)d_out);
}